// Preprocessing_24111946400026
// MI455X (gfx1250) — compile-verified
//
#include <hip/hip_runtime.h>
#include <stdint.h>
#include <float.h>
#include <math.h>

#define BS   256
#define LEN  5000

typedef __attribute__((ext_vector_type(2))) float v2f;
typedef __attribute__((ext_vector_type(8))) float v8f;

// db4 decomposition lowpass taps (dec_lo). dec_hi[k] = (-1)^(k+1) * dec_lo[k]
__device__ static constexpr float DLO[8] = {
  -0.010597401784997278f,  0.032883011666982945f,  0.030841381835986965f,
  -0.18703481171888114f,  -0.02798376941698385f,   0.6308807679295904f,
   0.7148465705525415f,    0.23037781330885523f
};

// lowpass pyramid lengths for L=5000: C = (L+7)/2 per level
__device__ static constexpr int PL[10]   = {5000,2503,1255,631,319,163,85,46,26,16};
__device__ static constexpr int POFF[10] = {0,5000,7503,8758,9389,9708,9871,9956,10002,10028};
#define PYR_TOTAL 10044

// ---------------------------------------------------------------------------
// CDNA5 async global->LDS row staging (gfx1250 ASYNCcnt path)
// ---------------------------------------------------------------------------
__device__ __forceinline__ void async_copy_row_to_lds(const float* __restrict__ gbase,
                                                      float* lds_dst, int nvec4) {
  // low 32 bits of a flat LDS pointer == LDS byte offset (LDS aperture rule)
  uint32_t lds0 = (uint32_t)(uintptr_t)(void*)lds_dst;
  for (int i = threadIdx.x; i < nvec4; i += BS) {
    uint32_t ldsoff = lds0 + (uint32_t)(i << 4);
    uint32_t voff   = (uint32_t)(i << 4);
    asm volatile("global_load_async_to_lds_b128 %0, %1, %2"
                 :: "v"(ldsoff), "v"(voff), "s"(gbase)
                 : "memory");
  }
  asm volatile("s_wait_asynccnt 0" ::: "memory");
  __syncthreads();
}

__device__ __forceinline__ float block_sum(float v, float* red) {
  const int tid = threadIdx.x;
  red[tid] = v;
  __syncthreads();
  for (int s = BS >> 1; s > 0; s >>= 1) {
    if (tid < s) red[tid] += red[tid + s];
    __syncthreads();
  }
  float r = red[0];
  __syncthreads();
  return r;
}

// One db4 DWT level with pywt/jnp 'reflect' extension:
//   lp[j] = sum_k src[ref(2j+1-k)] * DLO[k],  hp likewise with DHI.
// Returns per-thread partial sum of hp^2. Optionally stores hp (thresholded).
__device__ __forceinline__ float dwt_level(const float* __restrict__ src, int L,
                                           float* __restrict__ lp,
                                           float* __restrict__ hpdst,
                                           float sigma, bool thr) {
  const int C = (L + 7) >> 1;
  float acc = 0.f;
  for (int j = threadIdx.x; j < C; j += BS) {
    float lo = 0.f, hi = 0.f;
    const int b = 2 * j + 1;
#pragma unroll
    for (int k = 0; k < 8; ++k) {
      int q = b - k;
      q = (q < 0) ? -q : q;          // reflect (no edge repeat), left
      if (q >= L) q = 2 * L - 2 - q; // reflect, right
      const float v  = src[q];
      lo = fmaf(v, DLO[k], lo);
      const float hk = (k & 1) ? DLO[k] : -DLO[k]; // DHI[k]
      hi = fmaf(v, hk, hi);
    }
    lp[j] = lo;
    acc = fmaf(hi, hi, acc);
    if (hpdst) hpdst[j] = thr ? ((hi < sigma) ? 0.f : hi) : hi;
  }
  return acc;
}

// Inverse db4 step: out[i] = sum_m ca[m]*DLO[2m-i+1] + cd[m]*DHI[2m-i+1],
// m in [floor(i/2), floor(i/2)+3] clipped to [0,C). outL = 2C-6.
__device__ __forceinline__ void idwt_level(const float* __restrict__ ca,
                                           const float* __restrict__ cd,
                                           int C, float* __restrict__ out) {
  const int outL = 2 * C - 6;
  for (int i = threadIdx.x; i < outL; i += BS) {
    float r = 0.f;
    const int m0 = i >> 1;
    if (i & 1) { // taps 0,2,4,6 ; DHI[even] = -DLO[even]
#pragma unroll
      for (int u = 0; u < 4; ++u) {
        int m = m0 + u;
        if (m < C) {
          r = fmaf(ca[m], DLO[2 * u], r);
          r = fmaf(cd[m], -DLO[2 * u], r);
        }
      }
    } else {     // taps 1,3,5,7 ; DHI[odd] = +DLO[odd]
#pragma unroll
      for (int u = 0; u < 4; ++u) {
        int m = m0 + u;
        if (m < C) {
          r = fmaf(ca[m], DLO[2 * u + 1], r);
          r = fmaf(cd[m], DLO[2 * u + 1], r);
        }
      }
    }
    out[i] = r;
  }
}

// ---------------------------------------------------------------------------
// Kernel A: baseline removal + level-3 detail extraction. One block per row.
// ---------------------------------------------------------------------------
__global__ __launch_bounds__(BS) void k_baseline(const float* __restrict__ x,
                                                 float* __restrict__ s2out,
                                                 float* __restrict__ det3out) {
  __shared__ __align__(16) float sig[PYR_TOTAL]; // full lowpass pyramid
  __shared__ __align__(16) float bl[LEN];        // baseline (in-place recon)
  __shared__ float red[BS];
  __shared__ float ssds[9];
  __shared__ int   sdepth;
  const int tid = threadIdx.x;
  const int row = blockIdx.x;

  async_copy_row_to_lds(x + (size_t)row * LEN, sig, LEN / 4);

  // 9 DWT levels (t=0..8) with per-level detail SSD
#pragma unroll
  for (int t = 0; t < 9; ++t) {
    float part = dwt_level(sig + POFF[t], PL[t], sig + POFF[t + 1], nullptr, 0.f, false);
    float ssd = block_sum(part, red);
    if (tid == 0) ssds[t] = ssd;
  }
  __syncthreads();

  if (tid == 0) { // first strict local minimum of ssd, else 8 (loop break depth)
    int d = 8;
    for (int t = 2; t <= 8; ++t)
      if (ssds[t - 2] > ssds[t - 1] && ssds[t - 1] < ssds[t]) { d = t; break; }
    sdepth = d;
  }
  __syncthreads();
  const int depth = sdepth;

  // baseline: start from sigs[8], idwt_lo upward, switching to sigs[k] when depth<k+1
  for (int i = tid; i < PL[8]; i += BS) bl[i] = sig[POFF[8] + i];
  __syncthreads();

#pragma unroll
  for (int k = 7; k >= 0; --k) {
    const int Lk = PL[k], Lin = PL[k + 1];
    const float* sk = sig + POFF[k];
    float tmpv[20]; // register staging => in-place in bl[]
    int c = 0;
    for (int i = tid; i < Lk; i += BS) {
      float r = 0.f;
      const int m0 = i >> 1;
      if (i & 1) {
#pragma unroll
        for (int u = 0; u < 4; ++u) { int m = m0 + u; if (m < Lin) r = fmaf(bl[m], DLO[2 * u], r); }
      } else {
#pragma unroll
        for (int u = 0; u < 4; ++u) { int m = m0 + u; if (m < Lin) r = fmaf(bl[m], DLO[2 * u + 1], r); }
      }
      tmpv[c++] = (depth >= k + 1) ? r : sk[i];
    }
    __syncthreads();
    c = 0;
    for (int i = tid; i < Lk; i += BS) bl[i] = tmpv[c++];
    __syncthreads();
  }

  // s2 = x - baseline ; stash in LDS and in d_out (scratch for phase 2/3)
  float* orow = s2out + (size_t)row * LEN;
  for (int i = tid; i < LEN; i += BS) {
    float v = sig[i] - bl[i];
    sig[i]  = v;
    orow[i] = v;
  }
  __syncthreads();

  // 4-level DWT of s2; only level-3 detail (len 319) needed now (global median)
  float* d3 = det3out + (size_t)row * PL[4];
#pragma unroll
  for (int t = 0; t < 4; ++t) {
    dwt_level(sig + POFF[t], PL[t], sig + POFF[t + 1], (t == 3) ? d3 : nullptr, 0.f, false);
    __syncthreads();
  }
}

// ---------------------------------------------------------------------------
// Exact median of |det3| via 2-pass 16-bit radix select on float bit patterns.
// Prefix sums over the 256 per-thread partial counts are done on the matrix
// pipe: counts < 2^24, so f32 WMMA arithmetic is EXACT, and an inclusive
// prefix is W = V(16x16) x UpperTriangularOnes -- 4 chained v_wmma_f32_16x16x4_f32.
// ---------------------------------------------------------------------------
__global__ void k_hist_hi(const float* __restrict__ d, int n, uint32_t* __restrict__ hist) {
  int i = blockIdx.x * blockDim.x + threadIdx.x;
  if (i < n) {
    uint32_t b = __float_as_uint(d[i]) & 0x7fffffffu;
    atomicAdd(&hist[b >> 16], 1u);
  }
}

__global__ void k_hist_lo(const float* __restrict__ d, int n,
                          const uint32_t* __restrict__ sel, uint32_t* __restrict__ hist) {
  const uint32_t hi = sel[0];
  int i = blockIdx.x * blockDim.x + threadIdx.x;
  if (i < n) {
    uint32_t b = __float_as_uint(d[i]) & 0x7fffffffu;
    if ((b >> 16) == hi) atomicAdd(&hist[b & 0xffffu], 1u);
  }
}

// Inclusive prefix of part[256] -> pfx[256] using WMMA on wave 0 (EXEC all-1s).
// A = V chunks (16x4), B = U chunks (4x16, upper-tri ones), C/D accumulate.
// f32 layouts (wave32): A: lane&15 = M, lane-half selects K pair {0,1}/{2,3}
// across 2 VGPRs; B mirrors with N = lane&15; D: vgpr v -> M=v(+8), N = lane&15.
__device__ __forceinline__ void wmma_prefix256(const float* part, float* wrow, float* pfx) {
  const int tid = threadIdx.x;
  if (tid < 32) {
    const int half   = tid >> 4;
    const int lane16 = tid & 15;
    v8f acc = {};
#pragma unroll
    for (int c = 0; c < 4; ++c) {
      const int k0 = 4 * c + 2 * half;   // K index of the .x element
      v2f a, b;
      a.x = part[lane16 * 16 + k0];      // V[m][k0]
      a.y = part[lane16 * 16 + k0 + 1];  // V[m][k0+1]
      b.x = (k0     <= lane16) ? 1.f : 0.f; // U[k][n] = (k <= n)
      b.y = (k0 + 1 <= lane16) ? 1.f : 0.f;
      acc = __builtin_amdgcn_wmma_f32_16x16x4_f32(false, a, false, b,
                                                  (short)0, acc, false, false);
    }
#pragma unroll
    for (int v = 0; v < 8; ++v) wrow[(v + 8 * half) * 16 + lane16] = acc[v];
  }
  __syncthreads();
  // pfx[r*16+c] = sum_{r'<r} W[r'][15] + W[r][c]
  const int r = tid >> 4, cc = tid & 15;
  float s = wrow[r * 16 + cc];
  for (int rp = 0; rp < 15; ++rp)
    if (rp < r) s += wrow[rp * 16 + 15];
  pfx[tid] = s;
  __syncthreads();
}

// Find bin containing order statistic `target` in a 65536-bin histogram.
__device__ __forceinline__ void scan_find(const uint32_t* __restrict__ hist, float target,
                                          uint32_t* obin, uint32_t* ocnt) {
  __shared__ float part[256];
  __shared__ float wrow[256];
  __shared__ float pfx[256];
  __shared__ uint32_t res[2];
  const int tid = threadIdx.x;
  uint32_t s = 0;
  for (int j = 0; j < 256; ++j) s += hist[(tid << 8) + j];
  part[tid] = (float)s;   // exact: counts < 2^24
  __syncthreads();

  wmma_prefix256(part, wrow, pfx);

  const float lo = pfx[tid] - part[tid];      // exclusive prefix, exact
  if (lo <= target && target < pfx[tid]) {    // exactly one thread
    uint32_t acc = (uint32_t)lo;
    const uint32_t k = (uint32_t)target;
    int bin = tid << 8;
    for (int j = 0; j < 256; ++j) {
      uint32_t h = hist[(tid << 8) + j];
      if (acc + h > k) { bin = (tid << 8) + j; break; }
      acc += h;
    }
    res[0] = (uint32_t)bin;
    res[1] = acc;
  }
  __syncthreads();
  *obin = res[0];
  *ocnt = res[1];
}

__global__ void k_scan_hi(const uint32_t* __restrict__ hist, uint32_t ksel,
                          uint32_t* __restrict__ sel) {
  uint32_t bin, cnt;
  scan_find(hist, (float)ksel, &bin, &cnt);
  if (threadIdx.x == 0) { sel[0] = bin; sel[1] = cnt; }
}

__global__ void k_scan_lo(const uint32_t* __restrict__ hist, const uint32_t* __restrict__ sel,
                          uint32_t ksel, float* __restrict__ sigma) {
  uint32_t bin, cnt;
  scan_find(hist, (float)(ksel - sel[1]), &bin, &cnt);
  if (threadIdx.x == 0)
    sigma[0] = __uint_as_float((sel[0] << 16) | bin) / 0.6745f;
}

// ---------------------------------------------------------------------------
// Kernel C: thresholded 4-level DWT -> IDWT -> normalize. One block per row.
// ---------------------------------------------------------------------------
__global__ __launch_bounds__(BS) void k_denoise(float* __restrict__ io,
                                                const float* __restrict__ sigmap) {
  __shared__ __align__(16) float bufA[LEN];
  __shared__ __align__(16) float bufB[LEN];
  __shared__ float det0[2503];
  __shared__ float det1[1255];
  __shared__ float det2[631];
  __shared__ float det3[319];
  __shared__ float red[BS];
  const int tid = threadIdx.x;
  const int row = blockIdx.x;
  const float sigma = sigmap[0];
  float* rowp = io + (size_t)row * LEN;

  async_copy_row_to_lds(rowp, bufA, LEN / 4);

  // analysis with reference's signed threshold: d < sigma -> 0
  dwt_level(bufA, 5000, bufB, det0, sigma, true); __syncthreads();
  dwt_level(bufB, 2503, bufA, det1, sigma, true); __syncthreads();
  dwt_level(bufA, 1255, bufB, det2, sigma, true); __syncthreads();
  dwt_level(bufB,  631, bufA, det3, sigma, true); __syncthreads();

  // synthesis; ca implicitly trimmed to cd length via m<C clipping
  idwt_level(bufA, det3, 319,  bufB); __syncthreads(); // -> 632
  idwt_level(bufB, det2, 631,  bufA); __syncthreads(); // -> 1256
  idwt_level(bufA, det1, 1255, bufB); __syncthreads(); // -> 2504
  idwt_level(bufB, det0, 2503, bufA); __syncthreads(); // -> 5000

  // normalize: (x-mean)/std(ddof=1), nan_to_num
  float sm = 0.f;
  for (int i = tid; i < LEN; i += BS) sm += bufA[i];
  sm = block_sum(sm, red);
  const float mean = sm / (float)LEN;
  float sq = 0.f;
  for (int i = tid; i < LEN; i += BS) { float d = bufA[i] - mean; sq = fmaf(d, d, sq); }
  sq = block_sum(sq, red);
  const float sd  = sqrtf(sq / (float)(LEN - 1));
  const float inv = 1.0f / sd;
  for (int i = tid; i < LEN; i += BS) {
    float y = (bufA[i] - mean) * inv;
    if (isnan(y))      y = 0.f;
    else if (isinf(y)) y = (y > 0.f) ? FLT_MAX : -FLT_MAX;
    rowp[i] = y;
  }
}

// ---------------------------------------------------------------------------
extern "C" void kernel_launch(void* const* d_in, const int* in_sizes, int n_in,
                              void* d_out, int out_size, void* d_ws, size_t ws_size,
                              hipStream_t stream) {
  (void)n_in; (void)out_size; (void)ws_size;
  const float* x  = (const float*)d_in[0];
  float* out      = (float*)d_out;
  char*  ws       = (char*)d_ws;

  const int nrows = in_sizes[0] / LEN;          // 512*12 = 6144
  const int nd3   = nrows * 319;                // level-3 detail count
  const uint32_t ksel = (uint32_t)((nd3 - 1) / 2);

  float* det3 = (float*)ws;
  size_t off_hist = (((size_t)nd3 * sizeof(float)) + 255) & ~(size_t)255;
  uint32_t* hist_hi = (uint32_t*)(ws + off_hist);
  uint32_t* hist_lo = hist_hi + 65536;
  uint32_t* sel     = hist_lo + 65536;
  float*    sigma   = (float*)(sel + 2);

  hipMemsetAsync(hist_hi, 0, 2 * 65536 * sizeof(uint32_t), stream);

  k_baseline<<<nrows, BS, 0, stream>>>(x, out, det3);
  k_hist_hi <<<(nd3 + 255) / 256, 256, 0, stream>>>(det3, nd3, hist_hi);
  k_scan_hi <<<1, 256, 0, stream>>>(hist_hi, ksel, sel);
  k_hist_lo <<<(nd3 + 255) / 256, 256, 0, stream>>>(det3, nd3, sel, hist_lo);
  k_scan_lo <<<1, 256, 0, stream>>>(hist_lo, sel, ksel, sigma);
  k_denoise <<<nrows, BS, 0, stream>>>(out, sigma);
}